// ClassificationMPS_61168924230444
// MI455X (gfx1250) — compile-verified
//
#include <hip/hip_runtime.h>
#include <hip/hip_bf16.h>

// ClassificationMPS: N=256, B=1024, D_PHYS=2, D_BOND=32, C=10
// Only slices n=128 (nhalf) and n=255 (n-1) of the big einsum are used.
// One wave (32 lanes) handles a tile of 16 batches using fp32 WMMA
// (v_wmma_f32_16x16x4_f32), K-loop of 8 steps over the 32-long bond dim.

#define NTOT   256
#define BTOT   1024
#define DBOND  32
#define NCLS   10
#define NHALF  (NTOT / 2)
#define NLAST  (NTOT - 1)

typedef __attribute__((ext_vector_type(2))) float v2f;
typedef __attribute__((ext_vector_type(8))) float v8f;

__global__ __launch_bounds__(32)
void ClassificationMPS_kernel(const float* __restrict__ inputs,   // (256,1024,2)
                              const float* __restrict__ tensor,   // (256,32,32,2)
                              const float* __restrict__ Aout,     // (10,32,32)
                              float* __restrict__ out)            // (1024,10)
{
    __shared__ float sx255[16][2];       // inputs[255, b0+m, i]
    __shared__ float sx128[16][2];       // inputs[128, b0+m, i]
    __shared__ float st255[32][2];       // tensor[255, l, 0, i]
    __shared__ float sT128[2][32][32];   // tensor[128, l, r, i] -> [i][l][r]
    __shared__ float sAr[16][32];        // Ar[m][r]
    __shared__ float sP[2][16][32];      // P_i[m][l]
    __shared__ float sAl[16][32];        // Al[m][l]
    __shared__ float sG[16][NCLS * 32];  // G[m][o*32+l]

    const int lane = threadIdx.x;        // 0..31, one wave per block
    const int b0   = blockIdx.x * 16;    // batch tile base
    const int h    = lane >> 4;          // half-wave (0/1)
    const int mn   = lane & 15;          // M (or N) index within 16-tile

    // ---- stage the only data we actually need ----
    {
        const int m = lane >> 1, i = lane & 1;
        sx255[m][i] = inputs[(NLAST * BTOT + b0 + m) * 2 + i];
        sx128[m][i] = inputs[(NHALF * BTOT + b0 + m) * 2 + i];
        const int l = lane >> 1;
        st255[l][i]      = tensor[((NLAST * DBOND + l) * DBOND + 0) * 2 + i];
        st255[l + 16][i] = tensor[((NLAST * DBOND + l + 16) * DBOND + 0) * 2 + i];
    }
    for (int k = lane; k < 2 * 32 * 32; k += 32) {
        const int i = k & 1, r = (k >> 1) & 31, l = k >> 6;
        sT128[i][l][r] = tensor[((NHALF * DBOND + l) * DBOND + r) * 2 + i];
    }
    __syncthreads();

    // ---- Ar[m][r] = x255[m][0]*t255[r][0] + x255[m][1]*t255[r][1] ----
    for (int k = lane; k < 16 * 32; k += 32) {
        const int m = k >> 5, r = k & 31;
        sAr[m][r] = sx255[m][0] * st255[r][0] + sx255[m][1] * st255[r][1];
    }
    __syncthreads();

    // ---- P_i[m][l] = sum_r T128_i[l][r] * Ar[m][r]  (WMMA: M=16,N=32,K=32) ----
    for (int i = 0; i < 2; ++i) {
        for (int nt = 0; nt < 2; ++nt) {
            v8f acc = {};
#pragma unroll
            for (int s = 0; s < 8; ++s) {
                const int kb = 4 * s + 2 * h;           // this lane's K pair
                v2f a, b;
                a.x = sAr[mn][kb];                      // A[m][k], A = Ar tile
                a.y = sAr[mn][kb + 1];
                b.x = sT128[i][nt * 16 + mn][kb];       // B[k][n] = T128_i[l][r]
                b.y = sT128[i][nt * 16 + mn][kb + 1];
                acc = __builtin_amdgcn_wmma_f32_16x16x4_f32(
                        false, a, false, b, (short)0, acc, false, false);
            }
#pragma unroll
            for (int g = 0; g < 8; ++g)                 // D[g+8h][n]
                sP[i][g + 8 * h][nt * 16 + mn] = acc[g];
        }
    }

    // ---- G[m][o*32+l] = sum_r Aout[o][l][r] * Ar[m][r]  (M=16,N=320,K=32) ----
    for (int nt = 0; nt < (NCLS * 32) / 16; ++nt) {     // 20 N-tiles
        const int j = nt * 16 + mn;                     // j = o*32 + l
        const int o = j >> 5, l = j & 31;
        const float* ao = Aout + (o * DBOND + l) * DBOND;
        v8f acc = {};
#pragma unroll
        for (int s = 0; s < 8; ++s) {
            const int kb = 4 * s + 2 * h;
            v2f a, b;
            a.x = sAr[mn][kb];
            a.y = sAr[mn][kb + 1];
            b.x = ao[kb];                               // B[k][n] = Aout[o][l][r]
            b.y = ao[kb + 1];
            acc = __builtin_amdgcn_wmma_f32_16x16x4_f32(
                    false, a, false, b, (short)0, acc, false, false);
        }
#pragma unroll
        for (int g = 0; g < 8; ++g)
            sG[g + 8 * h][j] = acc[g];
    }
    __syncthreads();

    // ---- Al[m][l] = x128[m][0]*P0[m][l] + x128[m][1]*P1[m][l] ----
    for (int k = lane; k < 16 * 32; k += 32) {
        const int m = k >> 5, l = k & 31;
        sAl[m][l] = sx128[m][0] * sP[0][m][l] + sx128[m][1] * sP[1][m][l];
    }
    __syncthreads();

    // ---- out[b0+m][o] = sum_l Al[m][l] * G[m][o*32+l] ----
    for (int k = lane; k < 16 * NCLS; k += 32) {
        const int m = k / NCLS, o = k % NCLS;
        float acc = 0.0f;
#pragma unroll
        for (int l = 0; l < 32; ++l)
            acc += sAl[m][l] * sG[m][o * 32 + l];
        out[(b0 + m) * NCLS + o] = acc;
    }
}

extern "C" void kernel_launch(void* const* d_in, const int* in_sizes, int n_in,
                              void* d_out, int out_size, void* d_ws, size_t ws_size,
                              hipStream_t stream) {
    const float* inputs = (const float*)d_in[0];  // 256*1024*2
    const float* tensor = (const float*)d_in[1];  // 256*32*32*2
    const float* Aout   = (const float*)d_in[2];  // 10*32*32
    float* out = (float*)d_out;                   // 1024*10

    (void)in_sizes; (void)n_in; (void)out_size; (void)d_ws; (void)ws_size;

    ClassificationMPS_kernel<<<BTOT / 16, 32, 0, stream>>>(inputs, tensor, Aout, out);
}